// AtomicFunction_39805756899848
// MI455X (gfx1250) — compile-verified
//
#include <hip/hip_runtime.h>
#include <hip/hip_bf16.h>

typedef __attribute__((ext_vector_type(2))) float v2f;
typedef __attribute__((ext_vector_type(8))) float v8f;

#define DIM      256
#define XS_PITCH 260   // 256 + 4 pad: 260 % 64 == 4 -> bank-conflict-free wave loads
#define TILE_B   16

// ---------------------------------------------------------------------------
// Kernel 1: build symmetric pair-coefficient matrix S (256x256 f32) in d_ws.
//   c_k = (w0[k]+w1[k]) * w0[k] * w1[k];  S[i][j] = S[j][i] = c_k;  diag = 0.
// Each pair (i<j) appears exactly once -> scatter is race-free. 256 trailing
// threads zero the diagonal, so every S element is initialized each call.
// ---------------------------------------------------------------------------
__global__ __launch_bounds__(256) void build_s_kernel(
    const float* __restrict__ w,     // [2, NI] flat
    const int*   __restrict__ perm,  // [2, NI] flat: first NI = i, next NI = j
    float*       __restrict__ S,
    int NI) {
  int t = blockIdx.x * blockDim.x + threadIdx.x;
  if (t < NI) {
    int   i  = perm[t];
    int   j  = perm[NI + t];
    float w0 = w[t];
    float w1 = w[NI + t];
    float c  = (w0 + w1) * w0 * w1;
    S[i * DIM + j] = c;
    S[j * DIM + i] = c;
  } else if (t < NI + DIM) {
    int d = t - NI;
    S[d * DIM + d] = 0.0f;
  }
}

// ---------------------------------------------------------------------------
// Kernel 2: out[b] = 0.5 * x_b^T S x_b via WMMA f32 16x16x4.
// One workgroup (8 wave32s) per 16-batch tile. Each wave computes two 16x16
// tiles of Y = X_tile @ S, then reduces sum_n X[m][n]*Y[m][n] into LDS.
//
// Fragment layouts (CDNA5 ISA 7.12.2, 32-bit):
//   A (16x4):  lanes 0-15 -> M=lane, K={0,1} in vgpr{0,1};
//              lanes 16-31 -> M=lane-16, K={2,3}.
//   B (4x16):  lanes 0-15 -> N=lane, K={0,1}; lanes 16-31 -> N, K={2,3}.
//   C/D(16x16): vgpr r, lanes 0-15 -> M=r, N=lane; lanes 16-31 -> M=r+8.
// ---------------------------------------------------------------------------
__global__ __launch_bounds__(256) void quad_form_kernel(
    const float* __restrict__ X,   // [B, 256]
    const float* __restrict__ S,   // [256, 256]
    float*       __restrict__ out, // [B]
    int Bn) {
  __shared__ float Xs[TILE_B * XS_PITCH];
  __shared__ float outp[TILE_B];

  const int tid = threadIdx.x;
  const int b0  = blockIdx.x * TILE_B;

  // Stage the 16x256 X tile into LDS with b128 loads/stores.
  // 4096 floats = 1024 float4; 256 threads x 4. Row pitch 260 (==0 mod 4)
  // keeps every float4 slot 16B-aligned.
  const float4* Xt4 = (const float4*)(X + (size_t)b0 * DIM);
#pragma unroll
  for (int i = 0; i < 4; ++i) {
    int q = tid + i * 256;          // float4 index within the tile
    int r = q >> 6;                 // 64 float4 per row
    int c = (q & 63) << 2;          // column in floats
    *(float4*)&Xs[r * XS_PITCH + c] = Xt4[q];
  }
  if (tid < TILE_B) outp[tid] = 0.0f;
  __syncthreads();

  const int wave = tid >> 5;
  const int lane = tid & 31;
  const int half = lane >> 4;   // 0: lanes 0-15, 1: lanes 16-31
  const int lm   = lane & 15;

  const float* Xrow = &Xs[lm * XS_PITCH];   // A fragment source: row M=lm

#pragma unroll
  for (int ti = 0; ti < 2; ++ti) {
    const int n0 = (wave * 2 + ti) << 4;    // this wave's 16-column strip
    v8f acc = {};

#pragma unroll 4
    for (int kk = 0; kk < DIM; kk += 4) {
      const int k2 = kk + 2 * half;
      // A: X[M=lm][k2], X[M=lm][k2+1]  (8B-aligned pair -> ds_load_b64)
      v2f a;
      a.x = Xrow[k2];
      a.y = Xrow[k2 + 1];
      // B: S[k2][n0+lm], S[k2+1][n0+lm]
      const float* Sp = &S[(size_t)k2 * DIM + n0 + lm];
      v2f b;
      b.x = Sp[0];
      b.y = Sp[DIM];
      acc = __builtin_amdgcn_wmma_f32_16x16x4_f32(
          /*neg_a=*/false, a, /*neg_b=*/false, b,
          /*c_mod=*/(short)0, acc, /*reuse_a=*/false, /*reuse_b=*/false);
    }

    // Elementwise multiply by X tile and reduce over N into per-row sums.
    // Load all 8 X values into independent temps first so the compiler can
    // clause the ds_loads instead of serializing load->wait->mul->atomic.
    float xv[8];
#pragma unroll
    for (int r = 0; r < 8; ++r) {
      xv[r] = Xs[(r + 8 * half) * XS_PITCH + n0 + lm];
    }
#pragma unroll
    for (int r = 0; r < 8; ++r) {
      atomicAdd(&outp[r + 8 * half], acc[r] * xv[r]);   // ds_add_f32
    }
  }
  __syncthreads();

  if (tid < TILE_B && (b0 + tid) < Bn) {
    out[b0 + tid] = 0.5f * outp[tid];   // symmetric S double-counts pairs
  }
}

// ---------------------------------------------------------------------------
extern "C" void kernel_launch(void* const* d_in, const int* in_sizes, int n_in,
                              void* d_out, int out_size, void* d_ws, size_t ws_size,
                              hipStream_t stream) {
  const float* x    = (const float*)d_in[0];  // [B, 256] f32
  const float* w    = (const float*)d_in[1];  // [1, 2, NI] f32
  const int*   perm = (const int*)d_in[2];    // [2*NI] i32
  float*       out  = (float*)d_out;          // [B] f32
  float*       S    = (float*)d_ws;           // 256*256*4 = 256 KB scratch

  const int NI = in_sizes[1] / 2;             // 32640
  const int Bn = in_sizes[0] / DIM;           // 1024

  const int t1 = NI + DIM;                    // pairs + diagonal
  build_s_kernel<<<(t1 + 255) / 256, 256, 0, stream>>>(w, perm, S, NI);
  quad_form_kernel<<<(Bn + TILE_B - 1) / TILE_B, 256, 0, stream>>>(x, S, out, Bn);
}